// FNO1d_3307124818608
// MI455X (gfx1250) — compile-verified
//
#include <hip/hip_runtime.h>
#include <math.h>

#define NSP   8192
#define CCH   64
#define NMODE 16
#define BB    64
#define OUTC  10
#define HID   128

#define USE_ASYNC_LDS 1

typedef float v2f __attribute__((ext_vector_type(2)));
typedef float v4f __attribute__((ext_vector_type(4)));
typedef float v8f __attribute__((ext_vector_type(8)));

// D = A(16x4 f32) * B(4x16 f32) + C(16x16 f32), wave32 WMMA
__device__ __forceinline__ v8f wmma4(v2f a, v2f b, v8f c) {
  return __builtin_amdgcn_wmma_f32_16x16x4_f32(
      /*neg_a=*/false, a, /*neg_b=*/false, b,
      /*c_mod=*/(short)0, c, /*reuse_a=*/false, /*reuse_b=*/false);
}

// ---------------------------------------------------------------------------
// Twiddle tables: T[n][j] (8192 x 32) and Tt[j][n] (32 x 8192)
//   j <  16 : cos(2*pi*j*n/N) ; j >= 16 : -sin(2*pi*(j-16)*n/N)
// Forward rfft (modes 0..15):  X = h @ T -> [Xr | Xi]
// Inverse (scaling folded into Z): out = Z @ Tt
// ---------------------------------------------------------------------------
__global__ void twiddle_kernel(float* __restrict__ T, float* __restrict__ Tt) {
  int n = blockIdx.x * blockDim.x + threadIdx.x;
  if (n >= NSP) return;
  for (int k = 0; k < NMODE; ++k) {
    int m = (k * n) & (NSP - 1);   // arg reduction for accuracy
    float th = 6.28318530717958647692f * (float)m / (float)NSP;
    float s, c;
    __sincosf(th, &s, &c);
    T[(size_t)n * 32 + k]          = c;
    T[(size_t)n * 32 + 16 + k]     = -s;
    Tt[(size_t)k * NSP + n]        = c;
    Tt[(size_t)(16 + k) * NSP + n] = -s;
  }
}

// ---------------------------------------------------------------------------
// fc0: x(B,N,6) -> h(B,C,N)
// ---------------------------------------------------------------------------
__global__ void fc0_kernel(const float* __restrict__ x, const float* __restrict__ w,
                           const float* __restrict__ bias, float* __restrict__ h) {
  __shared__ float sw[CCH * 6];
  __shared__ float sb[CCH];
  int tid = threadIdx.x;
  for (int i = tid; i < CCH * 6; i += blockDim.x) sw[i] = w[i];
  for (int i = tid; i < CCH; i += blockDim.x) sb[i] = bias[i];
  __syncthreads();
  int b = blockIdx.y;
  int n = blockIdx.x * blockDim.x + tid;
  const float* xp = x + ((size_t)b * NSP + n) * 6;
  float xi[6];
#pragma unroll
  for (int i = 0; i < 6; ++i) xi[i] = xp[i];
  float* hp = h + (size_t)b * CCH * NSP + n;
  for (int c = 0; c < CCH; ++c) {
    float acc = sb[c];
#pragma unroll
    for (int i = 0; i < 6; ++i) acc += xi[i] * sw[c * 6 + i];
    hp[(size_t)c * NSP] = acc;
  }
}

// ---------------------------------------------------------------------------
// Forward DFT (16 modes): X(4096 x 32) = h(4096 x 8192) @ T(8192 x 32)
// 8 waves split K; cross-wave reduction via LDS. One 16-row M-tile per block.
// ---------------------------------------------------------------------------
__global__ void dft_kernel(const float* __restrict__ h, const float* __restrict__ T,
                           float* __restrict__ X) {
  __shared__ float red[8][32][16];
  int tid  = threadIdx.x;
  int wave = tid >> 5;
  int lane = tid & 31;
  int m0   = blockIdx.x * 16;
  int row  = lane & 15;
  int kp   = (lane >> 4) << 1;
  int col  = lane & 15;

  v8f accC = {};
  v8f accS = {};
  const float* ha = h + (size_t)(m0 + row) * NSP + kp;
  int kbeg = wave * (NSP / 8);
  int kend = kbeg + (NSP / 8);
  for (int k0 = kbeg; k0 < kend; k0 += 4) {
    v2f a;
    a.x = ha[k0];
    a.y = ha[k0 + 1];
    const float* tb = T + (size_t)(k0 + kp) * 32;
    v2f bc, bs;
    bc.x = tb[col];      bc.y = tb[32 + col];
    bs.x = tb[16 + col]; bs.y = tb[48 + col];
    accC = wmma4(a, bc, accC);
    accS = wmma4(a, bs, accS);
  }
#pragma unroll
  for (int r = 0; r < 8; ++r) {
    red[wave][lane][r]     = accC[r];
    red[wave][lane][r + 8] = accS[r];
  }
  __syncthreads();
  for (int v = tid; v < 512; v += 256) {
    int l = v >> 4, slot = v & 15;
    float s = 0.0f;
#pragma unroll
    for (int w = 0; w < 8; ++w) s += red[w][l][slot];
    int m = m0 + ((l >> 4) << 3) + (slot & 7);
    int c = l & 15;
    int j = (slot < 8) ? c : (16 + c);
    X[(size_t)m * 32 + j] = s;
  }
}

// ---------------------------------------------------------------------------
// Per-mode complex mix + irfft scaling folded into Z.
// ---------------------------------------------------------------------------
__global__ void mix_kernel(const float* __restrict__ X, const float* __restrict__ wr,
                           const float* __restrict__ wi, float* __restrict__ Z) {
  int k   = blockIdx.x;
  int m0  = blockIdx.y * 16;
  int n0  = blockIdx.z * 16;
  int lane = threadIdx.x & 31;
  int row = lane & 15;
  int kp  = (lane >> 4) << 1;
  int col = lane & 15;

  v8f rr = {}, ii = {}, ri = {}, ir = {};
  for (int k0 = 0; k0 < CCH; k0 += 4) {
    v2f ar, ai;
    const float* xp = X + ((size_t)(m0 + row) * CCH + k0 + kp) * 32 + k;
    ar.x = xp[0];  ar.y = xp[32];
    ai.x = xp[16]; ai.y = xp[48];
    v2f br, bi;
    const float* wrp = wr + ((size_t)(k0 + kp) * CCH + n0 + col) * NMODE + k;
    const float* wip = wi + ((size_t)(k0 + kp) * CCH + n0 + col) * NMODE + k;
    br.x = wrp[0]; br.y = wrp[CCH * NMODE];
    bi.x = wip[0]; bi.y = wip[CCH * NMODE];
    rr = wmma4(ar, br, rr);
    ii = wmma4(ai, bi, ii);
    ri = wmma4(ar, bi, ri);
    ir = wmma4(ai, br, ir);
  }
  float sr = (k == 0) ? (1.0f / NSP) : (2.0f / NSP);
#pragma unroll
  for (int r = 0; r < 8; ++r) {
    int b = m0 + ((lane >> 4) << 3) + r;
    int o = n0 + col;
    float* zp = Z + ((size_t)b * CCH + o) * 32;
    zp[k]      = (rr[r] - ii[r]) * sr;
    zp[16 + k] = (k == 0) ? 0.0f : (ri[r] + ir[r]) * (2.0f / NSP);
  }
}

// ---------------------------------------------------------------------------
// Fused layer: out = act( convW @ h + convB + Z @ Tt )
// Block = 4 waves x 128-column n-chunk. A-fragments (conv weights + Z rows)
// held in registers across 8 n-subtiles; B panels staged in LDS (async copy).
// ---------------------------------------------------------------------------
#define LSTR 132   // 128 + 4 pad (bank-conflict avoidance; keeps 16B alignment)

__global__ void layer_kernel(const float* __restrict__ h, const float* __restrict__ Z,
                             const float* __restrict__ Tt, const float* __restrict__ cw,
                             const float* __restrict__ cb, float* __restrict__ out,
                             int do_relu) {
  __shared__ float Bh[CCH][LSTR];   // h panel:  64 x 128
  __shared__ float Bt[32][LSTR];    // Tt panel: 32 x 128
  int tid = threadIdx.x;            // 128 threads, 4 waves
  int b   = blockIdx.y;
  int n0  = blockIdx.x * 128;
  const float* hb    = h + (size_t)b * CCH * NSP;
  const float* gbase = hb + n0;
  const float* tbase = Tt + n0;

#if USE_ASYNC_LDS
  for (int idx = tid; idx < CCH * 32; idx += 128) {
    int r = idx >> 5, c4 = (idx & 31) << 2;
    unsigned lo = (unsigned)(size_t)&Bh[r][c4];
    unsigned go = (unsigned)(((size_t)r * NSP + c4) * 4);
    asm volatile("global_load_async_to_lds_b128 %0, %1, %2"
                 :: "v"(lo), "v"(go), "s"(gbase) : "memory");
  }
  for (int idx = tid; idx < 32 * 32; idx += 128) {
    int r = idx >> 5, c4 = (idx & 31) << 2;
    unsigned lo = (unsigned)(size_t)&Bt[r][c4];
    unsigned go = (unsigned)(((size_t)r * NSP + c4) * 4);
    asm volatile("global_load_async_to_lds_b128 %0, %1, %2"
                 :: "v"(lo), "v"(go), "s"(tbase) : "memory");
  }
  asm volatile("s_wait_asynccnt 0" ::: "memory");
#else
  for (int idx = tid; idx < CCH * 32; idx += 128) {
    int r = idx >> 5, c4 = (idx & 31) << 2;
    *(v4f*)&Bh[r][c4] = *(const v4f*)(gbase + (size_t)r * NSP + c4);
  }
  for (int idx = tid; idx < 32 * 32; idx += 128) {
    int r = idx >> 5, c4 = (idx & 31) << 2;
    *(v4f*)&Bt[r][c4] = *(const v4f*)(tbase + (size_t)r * NSP + c4);
  }
#endif
  __syncthreads();

  int wave = tid >> 5, lane = tid & 31;
  int row = lane & 15, kp = (lane >> 4) << 1, col = lane & 15;
  int o0 = wave * 16;

  // Preload A fragments: conv weights (K=64 -> 16 frags), Z rows (K=32 -> 8)
  v2f aw[16];
#pragma unroll
  for (int k = 0; k < 16; ++k) {
    const float* ap = cw + (size_t)(o0 + row) * CCH + (k << 2) + kp;
    aw[k].x = ap[0]; aw[k].y = ap[1];
  }
  v2f az[8];
#pragma unroll
  for (int j = 0; j < 8; ++j) {
    const float* zp = Z + ((size_t)b * CCH + o0 + row) * 32 + (j << 2) + kp;
    az[j].x = zp[0]; az[j].y = zp[1];
  }
  float bia[8];
#pragma unroll
  for (int r = 0; r < 8; ++r) bia[r] = cb[o0 + ((lane >> 4) << 3) + r];

  for (int st = 0; st < 8; ++st) {
    int nb = st << 4;
    v8f acc = {};
#pragma unroll
    for (int k = 0; k < 16; ++k) {
      v2f bq;
      bq.x = Bh[(k << 2) + kp][nb + col];
      bq.y = Bh[(k << 2) + kp + 1][nb + col];
      acc = wmma4(aw[k], bq, acc);
    }
#pragma unroll
    for (int j = 0; j < 8; ++j) {
      v2f bq;
      bq.x = Bt[(j << 2) + kp][nb + col];
      bq.y = Bt[(j << 2) + kp + 1][nb + col];
      acc = wmma4(az[j], bq, acc);
    }
#pragma unroll
    for (int r = 0; r < 8; ++r) {
      int m = o0 + ((lane >> 4) << 3) + r;
      float v = acc[r] + bia[r];
      if (do_relu) v = fmaxf(v, 0.0f);
      out[((size_t)b * CCH + m) * NSP + n0 + nb + col] = v;
    }
  }
}

// ---------------------------------------------------------------------------
// Fused MLP head: relu(fc1) -> fc2 -> fc3 over a 64-column n-chunk.
// Weight fragments preloaded in registers; activations staged through LDS.
// ---------------------------------------------------------------------------
#define HSTR 68    // 64 + 4 pad

__global__ void head_kernel(const float* __restrict__ h,
                            const float* __restrict__ w1, const float* __restrict__ b1,
                            const float* __restrict__ w2, const float* __restrict__ b2,
                            const float* __restrict__ w3, const float* __restrict__ b3,
                            float* __restrict__ out) {
  __shared__ float Bh[CCH][HSTR];
  __shared__ float G[HID][HSTR];
  __shared__ float H2[CCH][HSTR];
  int tid = threadIdx.x;             // 256 threads, 8 waves
  int b   = blockIdx.y;
  int n0  = blockIdx.x * 64;
  const float* hb = h + (size_t)b * CCH * NSP;
  for (int idx = tid; idx < CCH * 16; idx += 256) {
    int r = idx >> 4, c4 = (idx & 15) << 2;
    *(v4f*)&Bh[r][c4] = *(const v4f*)(hb + (size_t)r * NSP + n0 + c4);
  }
  __syncthreads();

  int wave = tid >> 5, lane = tid & 31;
  int row = lane & 15, kp = (lane >> 4) << 1, col = lane & 15;
  int o1 = wave * 16;        // fc1 tile rows (0..127)
  int o2 = (wave & 3) * 16;  // fc2 tile rows (0..63)

  // Preload weight fragments
  v2f a1[16];
#pragma unroll
  for (int k = 0; k < 16; ++k) {
    const float* ap = w1 + (size_t)(o1 + row) * CCH + (k << 2) + kp;
    a1[k].x = ap[0]; a1[k].y = ap[1];
  }
  v2f a2[32];
  if (wave < 4) {
#pragma unroll
    for (int k = 0; k < 32; ++k) {
      const float* ap = w2 + (size_t)(o2 + row) * HID + (k << 2) + kp;
      a2[k].x = ap[0]; a2[k].y = ap[1];
    }
  }
  v2f a3[16];
  if (wave == 0) {
#pragma unroll
    for (int k = 0; k < 16; ++k) {
      a3[k].x = (row < OUTC) ? w3[(size_t)row * CCH + (k << 2) + kp] : 0.0f;
      a3[k].y = (row < OUTC) ? w3[(size_t)row * CCH + (k << 2) + kp + 1] : 0.0f;
    }
  }
  float bia1[8], bia2[8];
#pragma unroll
  for (int r = 0; r < 8; ++r) {
    bia1[r] = b1[o1 + ((lane >> 4) << 3) + r];
    bia2[r] = b2[o2 + ((lane >> 4) << 3) + r];
  }

  for (int st = 0; st < 4; ++st) {
    int nb = st << 4;
    // fc1 + relu (M=128, K=64): all 8 waves
    {
      v8f acc = {};
#pragma unroll
      for (int k = 0; k < 16; ++k) {
        v2f bq;
        bq.x = Bh[(k << 2) + kp][nb + col];
        bq.y = Bh[(k << 2) + kp + 1][nb + col];
        acc = wmma4(a1[k], bq, acc);
      }
#pragma unroll
      for (int r = 0; r < 8; ++r) {
        int m = o1 + ((lane >> 4) << 3) + r;
        G[m][nb + col] = fmaxf(acc[r] + bia1[r], 0.0f);
      }
    }
    __syncthreads();
    // fc2 (M=64, K=128): waves 0..3
    if (wave < 4) {
      v8f acc = {};
#pragma unroll
      for (int k = 0; k < 32; ++k) {
        v2f bq;
        bq.x = G[(k << 2) + kp][nb + col];
        bq.y = G[(k << 2) + kp + 1][nb + col];
        acc = wmma4(a2[k], bq, acc);
      }
#pragma unroll
      for (int r = 0; r < 8; ++r) {
        int m = o2 + ((lane >> 4) << 3) + r;
        H2[m][nb + col] = acc[r] + bia2[r];
      }
    }
    __syncthreads();
    // fc3 (M=10 padded to 16, K=64): wave 0 -> d_out[b][n][o]
    if (wave == 0) {
      v8f acc = {};
#pragma unroll
      for (int k = 0; k < 16; ++k) {
        v2f bq;
        bq.x = H2[(k << 2) + kp][nb + col];
        bq.y = H2[(k << 2) + kp + 1][nb + col];
        acc = wmma4(a3[k], bq, acc);
      }
#pragma unroll
      for (int r = 0; r < 8; ++r) {
        int m = ((lane >> 4) << 3) + r;
        if (m < OUTC) {
          out[((size_t)b * NSP + n0 + nb + col) * OUTC + m] = acc[r] + b3[m];
        }
      }
    }
  }
}

extern "C" void kernel_launch(void* const* d_in, const int* in_sizes, int n_in,
                              void* d_out, int out_size, void* d_ws, size_t ws_size,
                              hipStream_t stream) {
  const float* x     = (const float*)d_in[0];
  const float* fc0w  = (const float*)d_in[1];
  const float* fc0b  = (const float*)d_in[2];
  const float* swr   = (const float*)d_in[3];
  const float* swi   = (const float*)d_in[4];
  const float* convw = (const float*)d_in[5];
  const float* convb = (const float*)d_in[6];
  const float* w1    = (const float*)d_in[7];
  const float* b1    = (const float*)d_in[8];
  const float* w2    = (const float*)d_in[9];
  const float* b2    = (const float*)d_in[10];
  const float* w3    = (const float*)d_in[11];
  const float* b3    = (const float*)d_in[12];

  float* ws = (float*)d_ws;
  size_t off = 0;
  float* T  = ws + off; off += (size_t)NSP * 32;
  float* Tt = ws + off; off += (size_t)NSP * 32;
  float* hA = ws + off; off += (size_t)BB * CCH * NSP;
  float* hB = ws + off; off += (size_t)BB * CCH * NSP;
  float* X  = ws + off; off += (size_t)BB * CCH * 32;
  float* Z  = ws + off; off += (size_t)BB * CCH * 32;

  twiddle_kernel<<<NSP / 256, 256, 0, stream>>>(T, Tt);
  fc0_kernel<<<dim3(NSP / 256, BB), 256, 0, stream>>>(x, fc0w, fc0b, hA);

  float* cur = hA;
  float* nxt = hB;
  for (int l = 0; l < 4; ++l) {
    dft_kernel<<<(BB * CCH) / 16, 256, 0, stream>>>(cur, T, X);
    mix_kernel<<<dim3(NMODE, BB / 16, CCH / 16), 32, 0, stream>>>(
        X, swr + (size_t)l * CCH * CCH * NMODE, swi + (size_t)l * CCH * CCH * NMODE, Z);
    layer_kernel<<<dim3(NSP / 128, BB), 128, 0, stream>>>(
        cur, Z, Tt, convw + (size_t)l * CCH * CCH, convb + (size_t)l * CCH, nxt,
        (l < 3) ? 1 : 0);
    float* t = cur; cur = nxt; nxt = t;
  }

  head_kernel<<<dim3(NSP / 64, BB), 256, 0, stream>>>(
      cur, w1, b1, w2, b2, w3, b3, (float*)d_out);
}